// AdaptiveTemp_67843303408036
// MI455X (gfx1250) — compile-verified
//
#include <hip/hip_runtime.h>
#include <hip/hip_bf16.h>
#include <math.h>

// ---------------------------------------------------------------------------
// AdaptiveTemp for MI455X (gfx1250, wave32, WMMA)
//
// reference collapses to:
//   K = W^T W  (10x10, fixed)
//   per sample: logits = x@W + b ; s = softmax ; p = s*(1-C*nu)+nu ; n=sqrt(p)
//     A_ic = alpha [ e_i d_ic + f_i d_Mc - q_i s_c ],  q=e+f
//     G = A K A^T (symmetric)  => norm1*norminf = (max abs row sum)^2
//     rho^2 * alpha^2 = (1-C*nu)^2   (exact cancellation)
//   out = delta / (EPS * (1-C*nu)^2 * maxrowsum|bracket|)
// Heavy part: logits GEMM (25MB read) -> WMMA f32 16x16x4, LDS-staged tiles.
// ---------------------------------------------------------------------------

typedef __attribute__((ext_vector_type(2))) float v2f;
typedef __attribute__((ext_vector_type(8))) float v8f;

#define DD   3072
#define CC   10
#define MM   9
#define BB   2048
#define NUV  1e-7f
#define EPSV 0.1f
#define KCHUNK 64            // K elems staged per tile iteration
#define NCHUNK (DD / KCHUNK) // 48
#define NFRAG  (DD / 4)      // 768 WMMA k-steps total

// ---------------- kernel 1: K = W^T W  (10x10) ------------------------------
__global__ void at_wtw_kernel(const float* __restrict__ W, float* __restrict__ Kout) {
    __shared__ float red[256];
    const int c  = blockIdx.x / CC;
    const int c2 = blockIdx.x % CC;
    float s = 0.f;
    for (int d = threadIdx.x; d < DD; d += 256)
        s += W[d * CC + c] * W[d * CC + c2];
    red[threadIdx.x] = s;
    __syncthreads();
    for (int st = 128; st > 0; st >>= 1) {
        if ((int)threadIdx.x < st) red[threadIdx.x] += red[threadIdx.x + st];
        __syncthreads();
    }
    if (threadIdx.x == 0) Kout[blockIdx.x] = red[0];
}

// ---------------- kernel 2: pre-shuffle W into WMMA B-fragment layout -------
// B is 4x16 (KxN) f32, 2 VGPRs/lane:
//   lane<16 : V0 = W[k+0][n], V1 = W[k+1][n]
//   lane>=16: V0 = W[k+2][n], V1 = W[k+3][n],  n = lane&15 (zero-pad n>=10)
// Stored as Wfrag[t][lane*2 + {0,1}] so the wave does one coalesced b64 load.
__global__ void at_wfrag_kernel(const float* __restrict__ W, float* __restrict__ Wfrag) {
    const int tid = blockIdx.x * 256 + threadIdx.x;
    if (tid >= NFRAG * 32) return;
    const int t    = tid >> 5;
    const int lane = tid & 31;
    const int n    = lane & 15;
    const int kb   = t * 4 + (lane >> 4) * 2;
    const float v0 = (n < CC) ? W[kb * CC + n] : 0.f;
    const float v1 = (n < CC) ? W[(kb + 1) * CC + n] : 0.f;
    Wfrag[t * 64 + lane * 2 + 0] = v0;
    Wfrag[t * 64 + lane * 2 + 1] = v1;
}

// ---------------- kernel 3: fused logits-GEMM + per-sample norm -------------
// 128 threads = 4 waves; each wave owns a 16-sample tile. grid = 32 blocks.
__global__ void __launch_bounds__(128)
at_main_kernel(const float* __restrict__ x, const float* __restrict__ bias,
               const float* __restrict__ Kmat, const float* __restrict__ Wfrag,
               float* __restrict__ out) {
    __shared__ float sK[CC * CC];
    __shared__ float xtile[4][16][KCHUNK + 4];  // pad 68 -> conflict-free ds_load_b64
    __shared__ float ltile[4][16][17];          // logits transpose, padded

    const int w    = threadIdx.x >> 5;
    const int lane = threadIdx.x & 31;
    const int half = lane >> 4;      // 0: K0/K1 lanes, 1: K2/K3 lanes
    const int m    = lane & 15;      // A-matrix row / D-matrix col
    const int row0 = (blockIdx.x * 4 + w) * 16;

    for (int i = threadIdx.x; i < CC * CC; i += blockDim.x) sK[i] = Kmat[i];
    __syncthreads();

    v8f acc = {};
    for (int t64 = 0; t64 < NCHUNK; ++t64) {
        const int k0 = t64 * KCHUNK;
        if (t64 + 1 < NCHUNK)  // pull next chunk toward the caches early
            __builtin_prefetch(x + (size_t)(row0 + m) * DD + k0 + KCHUNK + half * 32, 0, 1);

        // stage 16x64 slab of x into LDS, fully coalesced float4 loads
        #pragma unroll
        for (int j = 0; j < 8; ++j) {
            const int f4 = j * 32 + lane;           // 0..255 float4 slots
            const int r  = f4 >> 4;
            const int cc4 = (f4 & 15) << 2;
            const float4 v = *(const float4*)(x + (size_t)(row0 + r) * DD + k0 + cc4);
            *(float4*)&xtile[w][r][cc4] = v;
        }
        __syncthreads();

        const float* wf = Wfrag + (size_t)t64 * 16 * 64;
        #pragma unroll
        for (int kk = 0; kk < 16; ++kk) {
            const int kr = (kk << 2) + (half << 1);
            v2f a;
            a.x = xtile[w][m][kr];
            a.y = xtile[w][m][kr + 1];
            const v2f bf = *(const v2f*)(wf + kk * 64 + lane * 2);
            acc = __builtin_amdgcn_wmma_f32_16x16x4_f32(false, a, false, bf,
                                                        (short)0, acc, false, false);
        }
        __syncthreads();
    }

    // D layout: VGPR r holds M = r + 8*half, N = lane&15 -> transpose via LDS
    #pragma unroll
    for (int r = 0; r < 8; ++r) ltile[w][r + half * 8][m] = acc[r];
    __syncthreads();

    if (lane < 16) {
        const int srow = row0 + lane;
        float lg[CC], mx = -1e30f;
        #pragma unroll
        for (int c = 0; c < CC; ++c) { lg[c] = ltile[w][lane][c] + bias[c]; mx = fmaxf(mx, lg[c]); }
        float ex[CC], sum = 0.f;
        #pragma unroll
        for (int c = 0; c < CC; ++c) { ex[c] = expf(lg[c] - mx); sum += ex[c]; }
        const float inv = 1.f / sum;
        float s[CC], nq[CC], sumn = 0.f;
        #pragma unroll
        for (int c = 0; c < CC; ++c) {
            s[c]  = ex[c] * inv;
            nq[c] = sqrtf(s[c] * (1.f - CC * NUV) + NUV);
            sumn += nq[c];
        }
        const float delta = 2.f * acosf(sumn * 0.3162277660168379f);  // 1/sqrt(10)

        // K-contractions shared by all G entries
        float Ks[CC], Kss = 0.f;
        #pragma unroll
        for (int c = 0; c < CC; ++c) {
            float t = 0.f;
            #pragma unroll
            for (int c2 = 0; c2 < CC; ++c2) t += sK[c * CC + c2] * s[c2];
            Ks[c] = t;
            Kss  += s[c] * t;
        }
        const float n9 = nq[MM], s9 = s[MM];
        float e[MM], f[MM], q[MM];
        #pragma unroll
        for (int i = 0; i < MM; ++i) {
            e[i] = (1.f - n9) * s[i] / nq[i];
            f[i] = nq[i] * s9 / n9;
            q[i] = e[i] + f[i];
        }
        const float K9  = Ks[MM];
        const float KMM = sK[CC * CC - 1];
        float maxrow = 0.f;
        #pragma unroll
        for (int i = 0; i < MM; ++i) {
            const float KiM = sK[i * CC + MM];
            const float eKi = e[i] * Ks[i];
            float row = 0.f;
            #pragma unroll
            for (int j = 0; j < MM; ++j) {
                const float G = e[i] * e[j] * sK[i * CC + j]
                              + e[i] * f[j] * KiM
                              + f[i] * e[j] * sK[j * CC + MM]
                              + f[i] * f[j] * KMM
                              - eKi * q[j]
                              - q[i] * (e[j] * Ks[j])
                              - (f[i] * q[j] + q[i] * f[j]) * K9
                              + q[i] * q[j] * Kss;
                row += fabsf(G);
            }
            maxrow = fmaxf(maxrow, row);
        }
        const float sc = 1.f - CC * NUV;        // rho^2 * alpha^2 cancellation
        out[srow] = delta / (EPSV * sc * sc * maxrow);
    }
}

// ---------------------------------------------------------------------------
extern "C" void kernel_launch(void* const* d_in, const int* in_sizes, int n_in,
                              void* d_out, int out_size, void* d_ws, size_t ws_size,
                              hipStream_t stream) {
    const float* data = (const float*)d_in[0];   // [2048,3,32,32]
    const float* W    = (const float*)d_in[1];   // [3072,10]
    const float* bias = (const float*)d_in[2];   // [10]
    float* out = (float*)d_out;                  // [2048]

    float* Kmat  = (float*)d_ws;                 // 100 floats
    float* Wfrag = (float*)d_ws + 128;           // 768*64 floats (196KB), 512B offset

    at_wtw_kernel  <<<CC * CC, 256, 0, stream>>>(W, Kmat);
    at_wfrag_kernel<<<(NFRAG * 32 + 255) / 256, 256, 0, stream>>>(W, Wfrag);
    at_main_kernel <<<BB / 64, 128, 0, stream>>>(data, bias, Kmat, Wfrag, out);
}